// TopKLocalFocus3D_61091614818306
// MI455X (gfx1250) — compile-verified
//
#include <hip/hip_runtime.h>
#include <hip/hip_bf16.h>

typedef __attribute__((ext_vector_type(2))) float v2f;
typedef __attribute__((ext_vector_type(8))) float v8f;

#define B_    4
#define C_    256
#define D_    32
#define H_    64
#define W_    64
#define K_    64
#define PROJ_ 512
#define CK_   16384      // C_*K_ (contraction length)
#define HW_   4096
#define DHW_  131072

// -------- workspace layout (in floats) --------
// [0, 65536)          vec  : (B_, CK_)     patch means
// [65536, 73728)      acc  : (PROJ_, 16)   split-K accumulator (cols 0..3 valid)
// [73728, 73984)      zpad : 256 zero floats (B-matrix padding rows)
// [73984, 74752)      cent : (B_*K_*3)     int centers
#define WS_VEC  0
#define WS_ACC  65536
#define WS_ZPAD 73728
#define WS_CENT 73984

// -------- output layout (d_out, floats) --------
// [0, 768)      centers (B,K,3)
// [768, 1024)   scores  (B,K)
// [1024, 3072)  feature (B,512)

// ============================================================
// 1) Top-K per batch: iterative argmax with strict lexicographic
//    "(score, -idx) < previous" predicate; tie -> lower index.
//    2 MB of scores -> fully L2-resident after the first pass.
// ============================================================
__global__ __launch_bounds__(1024)
void topk_kernel(const float* __restrict__ score, float* __restrict__ out,
                 int* __restrict__ cent)
{
    __shared__ float ss[1024];
    __shared__ int   si[1024];
    __shared__ float sps;
    __shared__ int   spi;

    const int b   = blockIdx.x;
    const int tid = threadIdx.x;
    const float* sm = score + (size_t)b * DHW_;

    float ps = __builtin_inff();   // previous selected score
    int   pi = -1;                 // previous selected index

    for (int k = 0; k < K_; ++k) {
        float best = -__builtin_inff();
        int   bidx = 0x7fffffff;
        for (int i = tid; i < DHW_; i += 1024) {
            float v = sm[i];
            bool below_prev = (v < ps) || (v == ps && i > pi);
            if (below_prev && (v > best || (v == best && i < bidx))) {
                best = v; bidx = i;
            }
        }
        ss[tid] = best; si[tid] = bidx;
        __syncthreads();
        for (int off = 512; off > 0; off >>= 1) {
            if (tid < off) {
                float s2 = ss[tid + off]; int i2 = si[tid + off];
                if (s2 > ss[tid] || (s2 == ss[tid] && i2 < si[tid])) {
                    ss[tid] = s2; si[tid] = i2;
                }
            }
            __syncthreads();
        }
        if (tid == 0) {
            float v = ss[0]; int idx = si[0];
            int z = idx >> 12, rem = idx & 4095, y = rem >> 6, x = rem & 63;
            int o = b * K_ + k;
            out[o * 3 + 0] = (float)z;
            out[o * 3 + 1] = (float)y;
            out[o * 3 + 2] = (float)x;
            out[B_ * K_ * 3 + o] = v;
            cent[o * 3 + 0] = z; cent[o * 3 + 1] = y; cent[o * 3 + 2] = x;
            sps = v; spi = idx;
        }
        __syncthreads();
        ps = sps; pi = spi;
        __syncthreads();
    }
}

// ============================================================
// 2) Masked 5x5x5 patch mean. One block per (k,b), one channel
//    per thread. Clamp+mask == just skip out-of-range voxels.
// ============================================================
__global__ __launch_bounds__(256)
void patch_kernel(const float* __restrict__ fm, const int* __restrict__ cent,
                  float* __restrict__ vec)
{
    const int k = blockIdx.x, b = blockIdx.y, c = threadIdx.x;
    const int o = b * K_ + k;
    const int z = cent[o * 3 + 0], y = cent[o * 3 + 1], x = cent[o * 3 + 2];
    const int z0 = max(z - 2, 0), y0 = max(y - 2, 0), x0 = max(x - 2, 0);
    const int z1 = min(z0 + 5, D_), y1 = min(y0 + 5, H_), x1 = min(x0 + 5, W_);
    const float cnt = (float)((z1 - z0) * (y1 - y0) * (x1 - x0));

    const float* base = fm + ((size_t)(b * C_ + c)) * DHW_;
    float s = 0.f;
    for (int iz = z0; iz < z1; ++iz)
        for (int iy = y0; iy < y1; ++iy) {
            const float* row = base + iz * HW_ + iy * W_;
            for (int ix = x0; ix < x1; ++ix) s += row[ix];
        }
    vec[(size_t)b * CK_ + (size_t)k * C_ + c] = s / cnt;
}

// ============================================================
// 3) Zero split-K accumulator + B-padding block (ws is poisoned).
//    <<<33,256>>> covers 8192 acc floats + 256 zpad floats.
// ============================================================
__global__ void zero_acc(float* __restrict__ acc)
{
    acc[blockIdx.x * 256 + threadIdx.x] = 0.f;
}

// ============================================================
// 4) Split-K GEMM with V_WMMA_F32_16X16X4_F32.
//    Out[p][b] = sum_k Wp[p][k] * vec[b][k].
//    grid.x = 32 P-tiles of 16 rows, grid.y = 64 K-chunks of 256.
//    A lane layout (16x4 f32): lanes 0-15 -> M=lane, K={k0,k0+1};
//    lanes 16-31 -> M=lane-16, K={k0+2,k0+3}. B symmetric with N.
//    Lanes with N>=4 read from a zeroed pad so the inner loop has
//    NO exec-mask divergence (WMMA wants EXEC all-1s anyway).
// ============================================================
__global__ __launch_bounds__(32)
void gemm_kernel(const float* __restrict__ Wp, const float* __restrict__ vec,
                 const float* __restrict__ zpad, float* __restrict__ acc)
{
    const int ptile  = blockIdx.x;          // 0..31
    const int kchunk = blockIdx.y;          // 0..63
    const int lane   = threadIdx.x;         // 0..31
    const int half   = lane >> 4;           // K sub-pair select
    const int l      = lane & 15;           // M (for A) / N (for B)

    const float* aPtr = Wp + (size_t)(ptile * 16 + l) * CK_ + kchunk * 256 + half * 2;
    // Uniform B pointer: padding lanes (N>=4) re-read the same 2 zero floats (zpad + half*2) every step.
    const float* bPtr = (l < B_)
        ? (vec + (size_t)l * CK_ + kchunk * 256 + half * 2)
        : (zpad + half * 2);

    v8f cacc = {};
#pragma unroll 4
    for (int kk = 0; kk < 256; kk += 4) {
        __builtin_prefetch((const void*)(aPtr + kk + 64), 0, 3); // near-scope prefetch
        v2f a  = *(const v2f*)(aPtr + kk);
        v2f bb = *(const v2f*)(bPtr + ((l < B_) ? kk : 0));
        // 8 args: (neg_a, A, neg_b, B, c_mod, C, reuse_a, reuse_b)
        cacc = __builtin_amdgcn_wmma_f32_16x16x4_f32(
            false, a, false, bb, (short)0, cacc, false, false);
    }

    // C/D layout: lanes 0-15: N=lane, VGPR i -> M=i ; lanes 16-31: M=8+i.
    const int mbase = ptile * 16 + (half ? 8 : 0);
#pragma unroll
    for (int i = 0; i < 8; ++i)
        atomicAdd(&acc[(mbase + i) * 16 + l], cacc[i]);
}

// ============================================================
// 5) Epilogue: feature[b][p] = acc[p][b] + bias[p]
// ============================================================
__global__ void final_kernel(const float* __restrict__ acc,
                             const float* __restrict__ bias,
                             float* __restrict__ out)
{
    const int t = blockIdx.x * 256 + threadIdx.x;   // 0..2047
    const int b = t >> 9;                           // 0..3
    const int p = t & 511;                          // 0..511
    out[1024 + t] = acc[p * 16 + b] + bias[p];
}

extern "C" void kernel_launch(void* const* d_in, const int* in_sizes, int n_in,
                              void* d_out, int out_size, void* d_ws, size_t ws_size,
                              hipStream_t stream)
{
    const float* fm   = (const float*)d_in[0];  // (4,256,32,64,64)
    const float* sm   = (const float*)d_in[1];  // (4,1,32,64,64)
    const float* Wp   = (const float*)d_in[2];  // (512,16384)
    const float* bias = (const float*)d_in[3];  // (512,)
    float* out = (float*)d_out;
    float* ws  = (float*)d_ws;

    float* vec  = ws + WS_VEC;
    float* acc  = ws + WS_ACC;
    float* zpad = ws + WS_ZPAD;
    int*   cent = (int*)(ws + WS_CENT);

    topk_kernel<<<B_, 1024, 0, stream>>>(sm, out, cent);
    patch_kernel<<<dim3(K_, B_), C_, 0, stream>>>(fm, cent, vec);
    zero_acc<<<33, 256, 0, stream>>>(acc);   // zeroes acc (8192) + zpad (256)
    gemm_kernel<<<dim3(32, 64), 32, 0, stream>>>(Wp, vec, zpad, acc);
    final_kernel<<<8, 256, 0, stream>>>(acc, bias, out);
}